// LengthRegulator_33122787787407
// MI455X (gfx1250) — compile-verified
//
#include <hip/hip_runtime.h>

// LengthRegulator for MI455X (gfx1250, wave32).
// Pure bandwidth op: ~268 MB out + <=268 MB in  =>  ~23 us floor @ 23.3 TB/s.
// No matrix math exists in this op, so no WMMA; we optimize the memory path:
// 128-bit vector loads/stores, non-temporal stores for the streamed output,
// LDS prefix-scan, and a uniform per-wave binary search.

typedef float v4f __attribute__((ext_vector_type(4)));

constexpr int B  = 32;    // batch
constexpr int T  = 512;   // phonemes
constexpr int D  = 512;   // feature dim (floats)
constexpr int L  = 4096;  // max_len (static in reference harness)
constexpr int D4 = D / 4; // float4 chunks per row = 128

// ---------------------------------------------------------------------------
// Kernel 1: per-batch duration round/clamp + inclusive prefix sum (LDS scan).
// One 512-thread block (16 wave32s) per batch.
// ---------------------------------------------------------------------------
__global__ void lr_scan_kernel(const float* __restrict__ durations,
                               int* __restrict__ csum,
                               float* __restrict__ len_out) {
    __shared__ int s[T];
    const int b = blockIdx.x;
    const int t = threadIdx.x;

    // jnp.round = round-half-even -> rintf under default RNE mode; clamp >= 0.
    float d = durations[b * T + t];
    int v = (int)fmaxf(rintf(d), 0.0f);
    s[t] = v;
    __syncthreads();

    // Hillis-Steele inclusive scan over 512 elements (9 steps).
    #pragma unroll
    for (int off = 1; off < T; off <<= 1) {
        int x   = s[t];
        int add = (t >= off) ? s[t - off] : 0;
        __syncthreads();
        s[t] = x + add;
        __syncthreads();
    }

    csum[b * T + t] = s[t];
    if (t == T - 1) {
        // output_lengths, concatenated after the expanded tensor in d_out.
        // Small exact integers (<= 5120) -> exact in float.
        len_out[b] = (float)s[t];
    }
}

// ---------------------------------------------------------------------------
// Kernel 2: frame gather. One wave32 per output row (b, p).
// Each lane moves 4 x float4 = 64 B; wave moves the full 2 KB row with
// 4 coalesced b128 loads + 4 non-temporal b128 stores.
// ---------------------------------------------------------------------------
__global__ void lr_gather_kernel(const float* __restrict__ hidden,
                                 const int* __restrict__ csum,
                                 float* __restrict__ out) {
    const int row  = blockIdx.x * 8 + (threadIdx.x >> 5); // 8 waves per block
    const int lane = threadIdx.x & 31;
    const int b = row >> 12;       // row / L  (L == 4096)
    const int p = row & (L - 1);   // row % L

    const int* __restrict__ cb = csum + b * T;  // 2 KB, L0/L2 resident
    const int total = cb[T - 1];
    const bool valid = (p < total);

    // searchsorted(csum, p, side='right'): first idx with csum[idx] > p.
    // Uniform across the wave -> scalar-friendly 9-step binary search.
    int lo = 0, hi = T;
    while (lo < hi) {
        int mid = (lo + hi) >> 1;
        if (cb[mid] <= p) lo = mid + 1; else hi = mid;
    }
    int idx = (lo < T - 1) ? lo : (T - 1);

    const v4f* __restrict__ src = (const v4f*)hidden + ((size_t)b * T + idx) * D4;
    v4f* __restrict__ dst       = (v4f*)out + (size_t)row * D4;

    #pragma unroll
    for (int k = 0; k < 4; ++k) {
        const int c = lane + 32 * k;
        v4f val;
        if (valid) {
            val = src[c];                       // global_load_b128 (temporal: rows reused via L2)
        } else {
            val = (v4f){0.0f, 0.0f, 0.0f, 0.0f};
        }
        __builtin_nontemporal_store(val, dst + c);  // global_store_b128 NT: write-once stream
    }
}

// ---------------------------------------------------------------------------
// Host launcher (graph-capture safe: only kernel launches on `stream`).
// d_in[0]=hidden (B,T,D) f32, d_in[1]=durations (B,T) f32, d_in[2]=max_len.
// d_out = [expanded (B,L,D) f32 | output_lengths (B,) as f32].
// d_ws: first B*T*4 = 64 KB used for int32 cumsum.
// ---------------------------------------------------------------------------
extern "C" void kernel_launch(void* const* d_in, const int* in_sizes, int n_in,
                              void* d_out, int out_size, void* d_ws, size_t ws_size,
                              hipStream_t stream) {
    const float* hidden    = (const float*)d_in[0];
    const float* durations = (const float*)d_in[1];
    // d_in[2] (max_len) is a device scalar; value is static (4096) per reference.

    float* out      = (float*)d_out;
    float* len_out  = out + (size_t)B * L * D;   // tail of d_out
    int*   csum     = (int*)d_ws;                // 64 KB scratch

    lr_scan_kernel<<<B, T, 0, stream>>>(durations, csum, len_out);

    const int rows = B * L;                      // 131072 output rows
    lr_gather_kernel<<<rows / 8, 256, 0, stream>>>(hidden, csum, out);
}